// DiceLoss_50500225466858
// MI455X (gfx1250) — compile-verified
//
#include <hip/hip_runtime.h>
#include <math.h>

typedef __attribute__((ext_vector_type(2))) float v2f;
typedef __attribute__((ext_vector_type(8))) float v8f;

#define BATCH 64
#define H 512
#define W 512
#define KS 31
#define PAD 15
#define ROWS_PER_BLOCK 16
#define LDS_PITCH 516        // multiple of 4 (16B-aligned rows for b128 stores);
                             // 516 mod 64 = 4 banks/row -> conflict-free column gathers
#define TILES 32             // W / 16
#define RBLOCKS 32           // H / 16

// ---------------------------------------------------------------------------
// Kernel 1: vertical 31-tap box sum, sliding window (2 adds/pixel), float4-
// vectorized: one thread per (image, 4 columns); every access is a coalesced
// 512B b128 row segment per wave.
// ---------------------------------------------------------------------------
__global__ void __launch_bounds__(256)
vbox_kernel(const float* __restrict__ mask, float* __restrict__ colsum) {
    int t = blockIdx.x * blockDim.x + threadIdx.x;     // 0 .. BATCH*W/4-1
    if (t >= BATCH * (W / 4)) return;
    int b  = t >> 7;                                   // / (W/4)
    int x4 = (t & (W / 4 - 1)) << 2;
    const float4* in  = (const float4*)(mask   + (size_t)b * H * W + x4);
    float4*       out = (float4*)      (colsum + (size_t)b * H * W + x4);
    const int RS = W / 4;                              // row stride in float4

    float4 s = make_float4(0.f, 0.f, 0.f, 0.f);
    #pragma unroll
    for (int y = 0; y < PAD + 1; ++y) {                // window at y=0: rows 0..15
        float4 v = in[(size_t)y * RS];
        s.x += v.x; s.y += v.y; s.z += v.z; s.w += v.w;
    }
    for (int y = 0; y < H; ++y) {
        out[(size_t)y * RS] = s;
        int add = y + PAD + 1;
        int sub = y - PAD;
        if (add < H) {
            float4 v = in[(size_t)add * RS];
            s.x += v.x; s.y += v.y; s.z += v.z; s.w += v.w;
        }
        if (sub >= 0) {
            float4 v = in[(size_t)sub * RS];
            s.x -= v.x; s.y -= v.y; s.z -= v.z; s.w -= v.w;
        }
    }
}

// ---------------------------------------------------------------------------
// Kernel 2: horizontal 31-tap box sum via banded 16x16 matmuls on the WMMA
// pipe, fused with weit/sigmoid/Dice partial reduction.
//   Y_i = W_L * X_{i-1} + Ones * X_i + W_R * X_{i+1}
//   W_L[r][c] = (c > r), W_R[r][c] = (c < r)   (window +/-15 spans 3 tiles)
// Each 16x16x16 product = 4 chained V_WMMA_F32_16X16X4_F32 (K=4 chunks).
// One wave per 16-row block; EXEC is all ones at every WMMA (uniform control).
// ---------------------------------------------------------------------------
__global__ void __launch_bounds__(32)
hbox_fuse_kernel(const float* __restrict__ pred,
                 const float* __restrict__ mask,
                 const float* __restrict__ colsum,
                 float* __restrict__ partials) {
    __shared__ float lds[ROWS_PER_BLOCK * LDS_PITCH];

    const int rb = blockIdx.x;             // row block 0..31
    const int b  = blockIdx.y;             // image 0..63
    const int l  = threadIdx.x;            // lane 0..31
    const int rowbase = rb * ROWS_PER_BLOCK;
    const size_t img  = (size_t)b * H * W;

    // Coalesced b128 stage of the 16x512 colsum block into padded LDS.
    for (int idx = l; idx < ROWS_PER_BLOCK * (W / 4); idx += 32) {
        int r  = idx >> 7;                 // / (W/4)
        int x4 = (idx & (W / 4 - 1)) << 2;
        *(float4*)&lds[r * LDS_PITCH + x4] =
            *(const float4*)&colsum[img + (size_t)(rowbase + r) * W + x4];
    }
    __syncthreads();

    const int half = l >> 4;               // lane half
    const int r16  = l & 15;

    // Constant banded A operands, per documented 16x4 f32 A layout
    // (lane half h holds K = 2h, 2h+1 in its two VGPRs).
    v2f aM; aM.x = 1.f; aM.y = 1.f;        // all-ones (middle tile)
    v2f aL[4], aR[4];
    #pragma unroll
    for (int k = 0; k < 4; ++k) {
        int cx = 4 * k + 2 * half;
        aL[k].x = (cx     > r16) ? 1.f : 0.f;
        aL[k].y = (cx + 1 > r16) ? 1.f : 0.f;
        aR[k].x = (cx     < r16) ? 1.f : 0.f;
        aR[k].y = (cx + 1 < r16) ? 1.f : 0.f;
    }

    const float* ldsrow = &lds[r16 * LDS_PITCH];   // this lane's N-row
    const float  invK2  = 1.0f / (float)(KS * KS);
    float accI = 0.f, accU = 0.f;

    for (int i = 0; i < TILES; ++i) {
        v8f acc = {};                                       // C = 0

        if (i > 0) {                                        // left neighbor tile
            #pragma unroll
            for (int k = 0; k < 4; ++k) {
                const float* p = ldsrow + 16 * (i - 1) + 4 * k + 2 * half;
                v2f bv; bv.x = p[0]; bv.y = p[1];
                acc = __builtin_amdgcn_wmma_f32_16x16x4_f32(
                        false, aL[k], false, bv, (short)0, acc, false, false);
            }
        }
        #pragma unroll
        for (int k = 0; k < 4; ++k) {                       // center tile (all-ones band)
            const float* p = ldsrow + 16 * i + 4 * k + 2 * half;
            v2f bv; bv.x = p[0]; bv.y = p[1];
            acc = __builtin_amdgcn_wmma_f32_16x16x4_f32(
                    false, aM, false, bv, (short)0, acc, false, false);
        }
        if (i < TILES - 1) {                                // right neighbor tile
            #pragma unroll
            for (int k = 0; k < 4; ++k) {
                const float* p = ldsrow + 16 * (i + 1) + 4 * k + 2 * half;
                v2f bv; bv.x = p[0]; bv.y = p[1];
                acc = __builtin_amdgcn_wmma_f32_16x16x4_f32(
                        false, aR[k], false, bv, (short)0, acc, false, false);
            }
        }

        // Consume: lane l, vgpr v holds (row = rowbase + r16, x = 16i + v + 8*half).
        int row = rowbase + r16;
        int xb  = 16 * i + 8 * half;
        const float4* mp = (const float4*)(mask + img + (size_t)row * W + xb);
        const float4* pp = (const float4*)(pred + img + (size_t)row * W + xb);
        float4 m0 = mp[0], m1 = mp[1];
        float4 p0 = pp[0], p1 = pp[1];
        float mv[8] = {m0.x, m0.y, m0.z, m0.w, m1.x, m1.y, m1.z, m1.w};
        float pv[8] = {p0.x, p0.y, p0.z, p0.w, p1.x, p1.y, p1.z, p1.w};

        #pragma unroll
        for (int v = 0; v < 8; ++v) {
            float avg  = acc[v] * invK2;
            float m    = mv[v];
            float weit = fmaf(5.f, fabsf(avg - m), 1.f);
            // fast sigmoid: one v_exp_f32 + one v_rcp_f32 (no IEEE divide chain)
            float sig  = __builtin_amdgcn_rcpf(1.f + __expf(-pv[v]));
            accI += sig * m * weit;
            accU += (sig + m) * weit;
        }
    }

    // Wave32 butterfly reduction, then deterministic per-block partial store.
    #pragma unroll
    for (int off = 16; off > 0; off >>= 1) {
        accI += __shfl_xor(accI, off, 32);
        accU += __shfl_xor(accU, off, 32);
    }
    if (l == 0) {
        int blk = b * RBLOCKS + rb;
        partials[2 * blk + 0] = accI;
        partials[2 * blk + 1] = accU;
    }
}

// ---------------------------------------------------------------------------
// Kernel 3: deterministic finalize. 64 threads, one per image: sum 32 block
// partials, compute wdiss, then tree-reduce the mean into d_out[0].
// ---------------------------------------------------------------------------
__global__ void __launch_bounds__(64)
finalize_kernel(const float* __restrict__ partials, float* __restrict__ out) {
    __shared__ float sh[64];
    int b = threadIdx.x;
    float I = 0.f, U = 0.f;
    #pragma unroll
    for (int rbk = 0; rbk < RBLOCKS; ++rbk) {
        I += partials[2 * (b * RBLOCKS + rbk) + 0];
        U += partials[2 * (b * RBLOCKS + rbk) + 1];
    }
    sh[b] = 1.f - (2.f * I + 0.5f) / (U + 0.5f);   // 64 divides total: keep exact
    __syncthreads();
    #pragma unroll
    for (int s = 32; s > 0; s >>= 1) {
        if (b < s) sh[b] += sh[b + s];
        __syncthreads();
    }
    if (b == 0) out[0] = sh[0] * (1.f / (float)BATCH);
}

// ---------------------------------------------------------------------------
extern "C" void kernel_launch(void* const* d_in, const int* in_sizes, int n_in,
                              void* d_out, int out_size, void* d_ws, size_t ws_size,
                              hipStream_t stream) {
    (void)in_sizes; (void)n_in; (void)out_size; (void)ws_size;
    const float* pred = (const float*)d_in[0];
    const float* mask = (const float*)d_in[1];

    // ws layout: [0, 16KB) per-block partials; [16KB, 16KB+67MB) colsum scratch
    float* partials = (float*)d_ws;
    float* colsum   = (float*)((char*)d_ws + 16384);

    vbox_kernel<<<(BATCH * (W / 4) + 255) / 256, 256, 0, stream>>>(mask, colsum);

    dim3 g2(RBLOCKS, BATCH);   // 32 row-blocks x 64 images, 1 wave each
    hbox_fuse_kernel<<<g2, 32, 0, stream>>>(pred, mask, colsum, partials);

    finalize_kernel<<<1, 64, 0, stream>>>(partials, (float*)d_out);
}